// GRUModel_43121471652214
// MI455X (gfx1250) — compile-verified
//
#include <hip/hip_runtime.h>
#include <hip/hip_bf16.h>

#define H_   2048
#define L_   6
#define G3H  (3 * H_)
#define OUTN 2048
#define KSG  2   // K-splits for the big parallel w_hh pass
#define KSI  8   // K-splits for the sequential w_ih passes (keep chip busy)
#define KSL  4   // K-splits for final linear

typedef float v2f __attribute__((ext_vector_type(2)));
typedef float v8f __attribute__((ext_vector_type(8)));
typedef float f4v __attribute__((ext_vector_type(4)));

// ---------------------------------------------------------------------------
// GEMV partial kernel: each wave computes 16 rows of y = W x over one K-slice
// using V_WMMA_F32_16X16X4_F32. Weights stream HBM->LDS via coalesced
// nontemporal B128 loads; LDS tile is re-read in the WMMA A-fragment layout.
// Writes partial[row * nks + ks] (deterministic two-stage reduction).
// ---------------------------------------------------------------------------
__global__ __launch_bounds__(256) void gemv16_partial_kernel(
    const float* __restrict__ W, long wStride,
    const float* __restrict__ x, long xStride,
    float* __restrict__ partial, long pStride,
    int ldw, int K, int nks) {
  __shared__ float xs[H_];
  __shared__ __align__(16) float tile[8][16 * 68];  // 68-float row pad: bank-conflict free

  const int lane = threadIdx.x & 31;
  const int wave = threadIdx.x >> 5;
  const int m    = lane & 15;   // WMMA M index (A rows) / N index (B, D cols)
  const int hi   = lane >> 4;   // half-wave: selects K=0,1 vs K=2,3 slots

  W       += (long)blockIdx.z * wStride;
  x       += (long)blockIdx.z * xStride;
  partial += (long)blockIdx.z * pStride;

  const int kslice = K / nks;
  const int kbeg   = blockIdx.y * kslice;
  const int r0     = (blockIdx.x * 8 + wave) * 16;

  // Stage the input-vector slice once per block (reused by all 8 waves).
  for (int i = threadIdx.x; i < kslice; i += 256) xs[i] = x[kbeg + i];
  __syncthreads();

  v8f acc0 = {};
  v8f acc1 = {};
  float* tw = tile[wave];

  for (int k0 = 0; k0 < kslice; k0 += 64) {
    const float* wbase = W + (long)r0 * ldw + kbeg + k0;
    // Cooperative, coalesced load of 16 rows x 64 floats into padded LDS tile.
#pragma unroll
    for (int i = 0; i < 8; ++i) {
      int f   = i * 32 + lane;
      int row = f >> 4;   // 0..15
      int c4  = f & 15;   // float4 column 0..15
      f4v v = __builtin_nontemporal_load(
          (const f4v*)(wbase + (long)row * ldw + c4 * 4));
      *(f4v*)(tw + row * 68 + c4 * 4) = v;
    }
    // 16 WMMA steps consume the 16x64 tile (K in chunks of 4).
#pragma unroll
    for (int kk = 0; kk < 16; ++kk) {
      // A fragment (16x4 f32): lane m -> W[m][k],W[m][k+1]; lane m+16 -> k+2,k+3
      const float* ap = tw + m * 68 + kk * 4 + 2 * hi;
      v2f A; A.x = ap[0]; A.y = ap[1];
      // B fragment (4x16 f32): column N=0 holds x; other columns zero.
      float xv0 = xs[k0 + kk * 4 + 2 * hi];
      float xv1 = xs[k0 + kk * 4 + 2 * hi + 1];
      v2f B; B.x = (m == 0) ? xv0 : 0.0f;
      B.y = (m == 0) ? xv1 : 0.0f;
      if (kk & 1)
        acc1 = __builtin_amdgcn_wmma_f32_16x16x4_f32(false, A, false, B,
                                                     (short)0, acc1, false, false);
      else
        acc0 = __builtin_amdgcn_wmma_f32_16x16x4_f32(false, A, false, B,
                                                     (short)0, acc0, false, false);
    }
  }

  // Column N=0 of D: lane 0 holds M=0..7 in v0..v7, lane 16 holds M=8..15.
  if (m == 0) {
    int base = r0 + hi * 8;
#pragma unroll
    for (int g = 0; g < 8; ++g)
      partial[(long)(base + g) * nks + blockIdx.y] = acc0[g] + acc1[g];
  }
}

// ---------------------------------------------------------------------------
// Per-layer gate math: reduce K-partials, add biases, sigmoid/tanh, blend.
// ---------------------------------------------------------------------------
__global__ __launch_bounds__(256) void layer_combine_kernel(
    const float* __restrict__ giP, const float* __restrict__ ghP,
    const float* __restrict__ b_ih, const float* __restrict__ b_hh,
    const float* __restrict__ hprev, float* __restrict__ hout) {
  int j = blockIdx.x * 256 + threadIdx.x;
  float gir = b_ih[j], giz = b_ih[j + H_], gin = b_ih[j + 2 * H_];
#pragma unroll
  for (int s = 0; s < KSI; ++s) {
    gir += giP[(long)j * KSI + s];
    giz += giP[(long)(j + H_) * KSI + s];
    gin += giP[(long)(j + 2 * H_) * KSI + s];
  }
  float ghr = b_hh[j], ghz = b_hh[j + H_], ghn = b_hh[j + 2 * H_];
#pragma unroll
  for (int s = 0; s < KSG; ++s) {
    ghr += ghP[(long)j * KSG + s];
    ghz += ghP[(long)(j + H_) * KSG + s];
    ghn += ghP[(long)(j + 2 * H_) * KSG + s];
  }
  float r = 1.0f / (1.0f + __expf(-(gir + ghr)));
  float z = 1.0f / (1.0f + __expf(-(giz + ghz)));
  float n = tanhf(gin + r * ghn);
  float h = hprev[j];
  hout[j] = (1.0f - z) * n + z * h;
}

// ---------------------------------------------------------------------------
// LayerNorm over H=2048, one block of 1024 threads (32 waves).
// ---------------------------------------------------------------------------
__global__ __launch_bounds__(1024) void ln_kernel(
    const float* __restrict__ h, const float* __restrict__ gamma,
    const float* __restrict__ beta, float* __restrict__ outv) {
  __shared__ float s1[1024];
  __shared__ float s2[1024];
  __shared__ float mu_s, rs_s;
  int t = threadIdx.x;
  float a = h[t];
  float b = h[t + 1024];
  s1[t] = a + b;
  s2[t] = a * a + b * b;
  __syncthreads();
  for (int off = 512; off > 0; off >>= 1) {
    if (t < off) { s1[t] += s1[t + off]; s2[t] += s2[t + off]; }
    __syncthreads();
  }
  if (t == 0) {
    float mu  = s1[0] * (1.0f / 2048.0f);
    float var = s2[0] * (1.0f / 2048.0f) - mu * mu;
    mu_s = mu;
    rs_s = rsqrtf(var + 1e-5f);
  }
  __syncthreads();
  float mu = mu_s, rs = rs_s;
  outv[t]        = (a - mu) * rs * gamma[t]        + beta[t];
  outv[t + 1024] = (b - mu) * rs * gamma[t + 1024] + beta[t + 1024];
}

// ---------------------------------------------------------------------------
// Final linear combine: pred = linP-reduced + lin_b + x_t[:OUT]
// ---------------------------------------------------------------------------
__global__ __launch_bounds__(256) void lin_combine_kernel(
    const float* __restrict__ linP, const float* __restrict__ lin_b,
    const float* __restrict__ x, float* __restrict__ pred) {
  int o = blockIdx.x * 256 + threadIdx.x;
  float s = lin_b[o] + x[o];
#pragma unroll
  for (int ks = 0; ks < KSL; ++ks) s += linP[(long)o * KSL + ks];
  pred[o] = s;
}

extern "C" void kernel_launch(void* const* d_in, const int* in_sizes, int n_in,
                              void* d_out, int out_size, void* d_ws, size_t ws_size,
                              hipStream_t stream) {
  (void)in_sizes; (void)n_in; (void)out_size; (void)ws_size;
  const float* x      = (const float*)d_in[0];  // (1,1,2048)
  const float* hidden = (const float*)d_in[1];  // (6,1,2048)
  const float* w_ih   = (const float*)d_in[2];  // (6,6144,2048)
  const float* w_hh   = (const float*)d_in[3];  // (6,6144,2048)
  const float* b_ih   = (const float*)d_in[4];  // (6,6144)
  const float* b_hh   = (const float*)d_in[5];  // (6,6144)
  const float* gamma  = (const float*)d_in[6];  // (2048,)
  const float* beta   = (const float*)d_in[7];  // (2048,)
  const float* lin_w  = (const float*)d_in[8];  // (2048,2048)
  const float* lin_b  = (const float*)d_in[9];  // (2048,)

  float* out = (float*)d_out;            // [0,2048)=pred, [2048,14336)=new_hidden
  float* hid_out = out + OUTN;           // per-layer h_new slots
  float* ws  = (float*)d_ws;
  float* ghP = ws;                                   // L*3H*KSG floats
  float* giP = ghP + (long)L_ * G3H * KSG;           // 3H*KSI floats
  float* lnO = giP + (long)G3H * KSI;                // H floats
  float* liP = lnO + H_;                             // H*KSL floats

  // 1) All h-side gates (independent of the layer chain): fully parallel.
  gemv16_partial_kernel<<<dim3(G3H / 128, KSG, L_), 256, 0, stream>>>(
      w_hh, (long)G3H * H_, hidden, H_, ghP, (long)G3H * KSG, H_, H_, KSG);

  // 2) Sequential layer chain: K-split gi GEMV + gate combine per layer.
  for (int l = 0; l < L_; ++l) {
    const float* inp = (l == 0) ? x : (hid_out + (long)(l - 1) * H_);
    gemv16_partial_kernel<<<dim3(G3H / 128, KSI, 1), 256, 0, stream>>>(
        w_ih + (long)l * G3H * H_, 0, inp, 0, giP, 0, H_, H_, KSI);
    layer_combine_kernel<<<dim3(H_ / 256), 256, 0, stream>>>(
        giP, ghP + (long)l * G3H * KSG, b_ih + (long)l * G3H,
        b_hh + (long)l * G3H, hidden + (long)l * H_, hid_out + (long)l * H_);
  }

  // 3) LayerNorm of top-layer output.
  ln_kernel<<<1, 1024, 0, stream>>>(hid_out + (long)(L_ - 1) * H_, gamma, beta, lnO);

  // 4) Final linear (2048x2048 GEMV) + residual.
  gemv16_partial_kernel<<<dim3(H_ / 128, KSL, 1), 256, 0, stream>>>(
      lin_w, 0, lnO, 0, liP, 0, H_, H_, KSL);
  lin_combine_kernel<<<dim3(OUTN / 256), 256, 0, stream>>>(liP, lin_b, x, out);
}